// GGNN_NoGRU_NoEdgeNets_1108101562484
// MI455X (gfx1250) — compile-verified
//
#include <hip/hip_runtime.h>
#include <hip/hip_bf16.h>
#include <math.h>

// Problem constants (match reference)
#define B_      4
#define N_      50000
#define D_      150
#define S_      3
#define E_      100000
#define PASSES_ 8
#define NODES_FLOATS (B_ * N_ * D_)   // 30,000,000 floats = 120 MB (fits in 192 MB L2)
#define EDGES_TOTAL  (B_ * S_ * E_)   // 1,200,000 edges

typedef float v2f __attribute__((ext_vector_type(2)));
typedef float v8f __attribute__((ext_vector_type(8)));

// ---------------------------------------------------------------------------
// x_next = x_cur  (vectorized init before scatter accumulation)
// ---------------------------------------------------------------------------
__global__ void copy_kernel(const float4* __restrict__ src,
                            float4* __restrict__ dst, int n4) {
    int i = blockIdx.x * blockDim.x + threadIdx.x;
    if (i < n4) dst[i] = src[i];
}

// ---------------------------------------------------------------------------
// x_next[dst] += x_cur[src] for every edge. One wave per edge; 32 lanes cover
// the 150-float feature row (contiguous 600B -> coalesced). Node state is L2
// resident, so gathers + global_atomic_add_f32 mostly hit L2.
// ---------------------------------------------------------------------------
__global__ void scatter_kernel(const float* __restrict__ xcur,
                               float* __restrict__ xnext,
                               const int* __restrict__ edges) {
    int wave = blockIdx.x * (blockDim.x >> 5) + (threadIdx.x >> 5);
    int lane = threadIdx.x & 31;
    if (wave >= EDGES_TOTAL) return;
    int b = wave / (S_ * E_);
    long long ebase = (long long)wave * 2;
    int dst = edges[ebase];       // edges[..., 0]
    int src = edges[ebase + 1];   // edges[..., 1]
    const float* srow = xcur + ((long long)b * N_ + src) * D_;
    float*       drow = xnext + ((long long)b * N_ + dst) * D_;
    for (int d = lane; d < D_; d += 32)
        atomicAdd(&drow[d], srow[d]);
}

// ---------------------------------------------------------------------------
// s[b][d] = sum_n x[b][n][d].  One block per (d, b), LDS tree reduction.
// ---------------------------------------------------------------------------
__global__ void readout_kernel(const float* __restrict__ x,
                               float* __restrict__ s) {
    int d = blockIdx.x;   // 0..149
    int b = blockIdx.y;   // 0..3
    __shared__ float red[256];
    float acc = 0.f;
    const float* base = x + (long long)b * N_ * D_ + d;
    for (int n = threadIdx.x; n < N_; n += 256)
        acc += base[(long long)n * D_];
    red[threadIdx.x] = acc;
    __syncthreads();
    for (int off = 128; off > 0; off >>= 1) {
        if (threadIdx.x < off) red[threadIdx.x] += red[threadIdx.x + off];
        __syncthreads();
    }
    if (threadIdx.x == 0) s[b * D_ + d] = red[0];
}

// ---------------------------------------------------------------------------
// Single-wave MLP head using V_WMMA_F32_16X16X4_F32.
//
// Key trick vs. naive version: all fragment loads are UNCONDITIONAL.
//  * Activations live in LDS padded to the full 16 WMMA rows (rows 4..15 = 0),
//    so the A fragment is one aligned ds_load_b64 with no lane predicate.
//  * W1 has 151 rows but K is padded to 152; the padded A column (k=151) is
//    zero, so B row 151 contributes nothing — clamp the row index to 150 and
//    load unconditionally (v_min instead of saveexec bracketing).
//  * W3 (80x10) is staged once into LDS zero-padded to 80x16 so layer 3's
//    N-guard is outside the WMMA loop.
// A-frag layout (f32 16x4): lanes 0-15: M=lane, K=kb..kb+1; lanes 16-31: K=kb+2..kb+3
// B-frag layout (f32 4x16): N = lane&15, same K split.
// D layout: VGPR r, lanes 0-15 -> M=r, N=lane  (rows 0..3 are the valid batch).
// ---------------------------------------------------------------------------
__device__ __forceinline__ float leaky(float x) { return x > 0.f ? x : 0.01f * x; }

__global__ __launch_bounds__(32) void mlp_kernel(
    const float* __restrict__ s,  const float* __restrict__ ptype,
    const float* __restrict__ W1, const float* __restrict__ b1,
    const float* __restrict__ W2, const float* __restrict__ b2,
    const float* __restrict__ W3, const float* __restrict__ b3,
    float* __restrict__ out) {
    __shared__ float h0[16][152];   // [16, 152] zero-padded (valid: [4, 151])
    __shared__ float h1[16][80];
    __shared__ float h2[16][80];
    __shared__ float w3[80][16];    // W3 zero-padded N: 10 -> 16

    const int lane = threadIdx.x;
    const int m    = lane & 15;
    const int koff = (lane >> 4) << 1;   // 0 for lanes 0-15, 2 for lanes 16-31

    // ---- Zero all padded LDS tiles once ----
    for (int idx = lane; idx < 16 * 152; idx += 32) ((float*)h0)[idx] = 0.f;
    for (int idx = lane; idx < 16 * 80;  idx += 32) ((float*)h1)[idx] = 0.f;
    for (int idx = lane; idx < 16 * 80;  idx += 32) ((float*)h2)[idx] = 0.f;

    // ---- Stage W3 zero-padded (80x10 -> 80x16) ----
    for (int idx = lane; idx < 80 * 16; idx += 32) {
        int r = idx >> 4, c = idx & 15;
        int cc = c < 10 ? c : 9;                      // clamp: load in-bounds
        float v = W3[r * 10 + cc];
        w3[r][c] = (c < 10) ? v : 0.f;
    }

    // ---- Build h0 rows 0..3 (readout features + problem_type) ----
    for (int idx = lane; idx < B_ * 151; idx += 32) {
        int b = idx / 151, k = idx % 151;
        float v;
        if (k < D_) {
            float l = logf(s[b * D_ + k]);
            if (l != l) l = 0.f;          // nan -> 0
            v = fmaxf(l, 0.f);            // relu (also -inf -> 0)
        } else {
            v = ptype[b];                 // k == 150
        }
        h0[b][k] = v;
    }
    __syncthreads();

    // ---- Layer 1: [16,152] x [151,80] -> h1 (leaky relu) ---------------
    for (int t = 0; t < 5; ++t) {
        v8f acc = {0.f, 0.f, 0.f, 0.f, 0.f, 0.f, 0.f, 0.f};
        const int n = t * 16 + (lane & 15);           // < 80 always
        for (int kk = 0; kk < 38; ++kk) {
            const int kb = kk * 4 + koff;
            const v2f a = *reinterpret_cast<const v2f*>(&h0[m][kb]);  // ds_load_b64
            int k1 = kb + 1; if (k1 > 150) k1 = 150;  // A col 151 is 0 -> exact
            v2f bb;
            bb.x = W1[kb * 80 + n];
            bb.y = W1[k1 * 80 + n];
            acc = __builtin_amdgcn_wmma_f32_16x16x4_f32(
                false, a, false, bb, (short)0, acc, false, false);
        }
        if (lane < 16) {
            #pragma unroll
            for (int r = 0; r < B_; ++r)
                h1[r][n] = leaky(acc[r] + b1[n]);
        }
        __syncthreads();
    }

    // ---- Layer 2: [16,80] x [80,80] -> h2 (leaky relu) -----------------
    for (int t = 0; t < 5; ++t) {
        v8f acc = {0.f, 0.f, 0.f, 0.f, 0.f, 0.f, 0.f, 0.f};
        const int n = t * 16 + (lane & 15);
        for (int kk = 0; kk < 20; ++kk) {
            const int kb = kk * 4 + koff;
            const v2f a = *reinterpret_cast<const v2f*>(&h1[m][kb]);  // ds_load_b64
            v2f bb;
            bb.x = W2[kb * 80 + n];
            bb.y = W2[(kb + 1) * 80 + n];
            acc = __builtin_amdgcn_wmma_f32_16x16x4_f32(
                false, a, false, bb, (short)0, acc, false, false);
        }
        if (lane < 16) {
            #pragma unroll
            for (int r = 0; r < B_; ++r)
                h2[r][n] = leaky(acc[r] + b2[n]);
        }
        __syncthreads();
    }

    // ---- Layer 3: [16,80] x [80,16] -> out[4,10] -----------------------
    {
        v8f acc = {0.f, 0.f, 0.f, 0.f, 0.f, 0.f, 0.f, 0.f};
        const int n = lane & 15;
        for (int kk = 0; kk < 20; ++kk) {
            const int kb = kk * 4 + koff;
            const v2f a  = *reinterpret_cast<const v2f*>(&h2[m][kb]); // ds_load_b64
            const v2f bb = { w3[kb][n], w3[kb + 1][n] };
            acc = __builtin_amdgcn_wmma_f32_16x16x4_f32(
                false, a, false, bb, (short)0, acc, false, false);
        }
        if (lane < 10) {
            #pragma unroll
            for (int r = 0; r < B_; ++r)
                out[r * 10 + lane] = acc[r] + b3[lane];
        }
    }
}

// ---------------------------------------------------------------------------
extern "C" void kernel_launch(void* const* d_in, const int* in_sizes, int n_in,
                              void* d_out, int out_size, void* d_ws, size_t ws_size,
                              hipStream_t stream) {
    const float* nodes = (const float*)d_in[0];
    const int*   edges = (const int*)  d_in[1];
    const float* ptype = (const float*)d_in[2];
    const float* W1 = (const float*)d_in[3];
    const float* b1 = (const float*)d_in[4];
    const float* W2 = (const float*)d_in[5];
    const float* b2 = (const float*)d_in[6];
    const float* W3 = (const float*)d_in[7];
    const float* b3 = (const float*)d_in[8];
    float* out = (float*)d_out;

    // Workspace: two ping-pong node buffers (120 MB each) + readout vector.
    float* buf0 = (float*)d_ws;
    float* buf1 = buf0 + NODES_FLOATS;
    float* sbuf = buf1 + NODES_FLOATS;

    const int n4 = NODES_FLOATS / 4;                 // 7.5M float4
    dim3 cgrid((n4 + 255) / 256);
    dim3 sgrid(EDGES_TOTAL / 8);                     // 8 waves (edges) per 256-thr block

    const float* cur = nodes;
    for (int p = 0; p < PASSES_; ++p) {
        float* next = (p & 1) ? buf1 : buf0;
        copy_kernel<<<cgrid, 256, 0, stream>>>((const float4*)cur, (float4*)next, n4);
        scatter_kernel<<<sgrid, 256, 0, stream>>>(cur, next, edges);
        cur = next;
    }

    readout_kernel<<<dim3(D_, B_), 256, 0, stream>>>(cur, sbuf);
    mlp_kernel<<<1, 32, 0, stream>>>(sbuf, ptype, W1, b1, W2, b2, W3, b3, out);
}